// PretrainGGNN_74517682586461
// MI455X (gfx1250) — compile-verified
//
#include <hip/hip_runtime.h>
#include <hip/hip_bf16.h>

// GGNN: N=50000 nodes, E=800000 edges, D=64, L=5 steps.
#define NN      50000
#define EE      800000
#define DD      64
#define LL      5
#define TILES   (NN / 16)          // 3125 row tiles, exact
#define GRID_T  ((TILES + 7) / 8)  // 8 waves (of 32) per 256-thread block
#define LROW    68                 // padded LDS row stride: 16B-aligned rows,
                                   // lane bank step = 4 -> conflict-free b64

typedef float v2f __attribute__((ext_vector_type(2)));
typedef float v8f __attribute__((ext_vector_type(8)));

// f32 WMMA: D(16x16,f32) = A(16x4,f32) * B(4x16,f32) + C
__device__ __forceinline__ v8f wmma4(v2f a, v2f b, v8f c) {
  return __builtin_amdgcn_wmma_f32_16x16x4_f32(false, a, false, b, (short)0, c,
                                               false, false);
}

// gfx1250 async global->LDS b128 copy (ASYNCcnt-tracked, no VGPR round-trip).
// The LDS destination offset is derived from the pointer via ptrtoint: the low
// 32 bits of a generic pointer to LDS are the wave-relative LDS byte offset
// (ISA 10.2 aperture rules). The ptrtoint also marks the __shared__ object as
// address-taken, so the "memory" clobbers here and on the wait are assumed to
// write it — this is what keeps the later ds_load reads alive and ordered
// (round 2 showed they fold to undef otherwise; "m" operands crash ISel).
__device__ __forceinline__ void async_ld128(float* dst_lds, const void* g) {
  const unsigned off = (unsigned)(uintptr_t)dst_lds;  // LDS byte offset
  asm volatile("global_load_async_to_lds_b128 %0, %1, off"
               :: "v"(off), "v"(g)
               : "memory");
}
__device__ __forceinline__ void wait_async0() {
  asm volatile("s_wait_asynccnt 0x0" ::: "memory");
}

// ---------------------------------------------------------------------------
// Kernel 1: m = h @ W  (per-step 64x64 weight), and zero agg for the scatter.
// One wave per 16-row tile; computes all 4 16-col tiles (K = 64 = 16 x 4).
// ---------------------------------------------------------------------------
__global__ void ggnn_msg_gemm(const float* __restrict__ h,
                              const float* __restrict__ W,  // [64][64] (k, d)
                              float* __restrict__ m,
                              float* __restrict__ agg) {
  const int wave = (blockIdx.x * blockDim.x + threadIdx.x) >> 5;
  if (wave >= TILES) return;                 // wave-uniform: EXEC stays all-1s
  const int lane = threadIdx.x & 31;
  const int r0   = wave * 16;
  const int n    = lane & 15;                // row (A) / col (B,C)
  const int half = lane >> 4;
  const int kb   = half << 1;                // K offset 0 or 2 within k-tile

  // Zero agg tile (16 rows x 64 cols = 256 float4, 8 per lane), coalesced.
  {
    float4  z4 = make_float4(0.f, 0.f, 0.f, 0.f);
    float4* at = (float4*)(agg + (size_t)r0 * DD);
#pragma unroll
    for (int i = 0; i < 8; ++i) at[lane + 32 * i] = z4;
  }

  // A fragments: a[kt] = h[r0+n][kt*4 + kb + {0,1}]
  v2f a[16];
#pragma unroll
  for (int kt = 0; kt < 16; ++kt) {
    const float* p = h + (size_t)(r0 + n) * DD + kt * 4 + kb;
    a[kt].x = p[0];
    a[kt].y = p[1];
  }

#pragma unroll
  for (int ct = 0; ct < 4; ++ct) {
    v8f acc = {0.f, 0.f, 0.f, 0.f, 0.f, 0.f, 0.f, 0.f};
#pragma unroll
    for (int kt = 0; kt < 16; ++kt) {
      const float* p = W + (size_t)(kt * 4 + kb) * DD + ct * 16 + n;
      v2f b;
      b.x = p[0];
      b.y = p[DD];
      acc = wmma4(a[kt], b, acc);
    }
#pragma unroll
    for (int i = 0; i < 8; ++i) {
      const int row = r0 + i + half * 8;
      m[(size_t)row * DD + ct * 16 + n] = acc[i];
    }
  }
}

// ---------------------------------------------------------------------------
// Kernel 2: agg[dst] += m[src] over all edges. Half-wave (16 lanes) per edge,
// float4 gathers + hardware global_atomic_add_f32 (resolves in L2). This is
// the roofline limiter (~51M L2-resident f32 atomics per step).
// ---------------------------------------------------------------------------
__global__ void ggnn_scatter(const int* __restrict__ ei,   // [2][E]
                             const float* __restrict__ m,
                             float* __restrict__ agg) {
  const long long tid = (long long)blockIdx.x * blockDim.x + threadIdx.x;
  if (tid >= (long long)EE * 16) return;
  const int e = (int)(tid >> 4);
  const int c = (int)(tid & 15);
  const int src = ei[e];
  const int dst = ei[EE + e];
  const float4 v = ((const float4*)(m + (size_t)src * DD))[c];
  float* d = agg + (size_t)dst * DD + c * 4;
  unsafeAtomicAdd(d + 0, v.x);
  unsafeAtomicAdd(d + 1, v.y);
  unsafeAtomicAdd(d + 2, v.z);
  unsafeAtomicAdd(d + 3, v.w);
}

// ---------------------------------------------------------------------------
// Kernel 3: fused GRU cell.  gi = agg @ w_ih^T + b_ih, gh = h @ w_hh^T + b_hh,
// gates (r,z,n) then h' = (1-z)*n + z*h.  One wave per 16-row tile.
// Per 16-col output slice (dt) the block stages the 48+48 needed weight rows
// into LDS via async global->LDS b128 copies (padded stride LROW=68: 16B-
// aligned chunks, conflict-free ds_load_b64 B-fragment reads).
// NOTE: h / hout may alias (in-place update) -> no restrict on them. Tail
// waves clamp to the last tile, join staging + barriers, but do not store.
// ---------------------------------------------------------------------------
__global__ void __launch_bounds__(256) ggnn_gru(
    const float* __restrict__ agg,
    const float* h,
    const float* __restrict__ w_ih,  // [192][64]
    const float* __restrict__ w_hh,  // [192][64]
    const float* __restrict__ b_ih,  // [192]
    const float* __restrict__ b_hh,  // [192]
    float* hout) {
  // [s][ (g*16 + col) * LROW + k ] for s in {ih, hh}: 2*48*68*4B = 25.5 KB
  __shared__ float lw[2][48 * LROW];

  const int  tid    = threadIdx.x;
  const int  wave0  = (blockIdx.x * blockDim.x + tid) >> 5;
  const bool active = (wave0 < TILES);                 // wave-uniform
  const int  wave   = active ? wave0 : (TILES - 1);
  const int  lane   = tid & 31;
  const int  r0     = wave * 16;
  const int  n      = lane & 15;
  const int  half   = lane >> 4;
  const int  kb     = half << 1;

  // A fragments for both GEMM inputs (agg and h), K = 64.
  v2f aA[16], aH[16];
#pragma unroll
  for (int kt = 0; kt < 16; ++kt) {
    const float* pa = agg + (size_t)(r0 + n) * DD + kt * 4 + kb;
    aA[kt].x = pa[0];
    aA[kt].y = pa[1];
    const float* ph = h + (size_t)(r0 + n) * DD + kt * 4 + kb;
    aH[kt].x = ph[0];
    aH[kt].y = ph[1];
  }

#pragma unroll
  for (int dt = 0; dt < 4; ++dt) {
    __syncthreads();  // previous slice's LDS readers are done

    // Stage: per matrix 48 rows x 64 floats = 768 float4; 3 per thread.
#pragma unroll
    for (int s = 0; s < 2; ++s) {
      const float* wsrc = s ? w_hh : w_ih;
#pragma unroll
      for (int q = 0; q < 3; ++q) {
        const int idx = tid + 256 * q;      // 0..767
        const int row = idx >> 4;           // 0..47  (= g*16 + col)
        const int c4  = idx & 15;           // float4 within the row
        const int j   = (row >> 4) * DD + dt * 16 + (row & 15);
        const int fo  = (s * 48 + row) * LROW + c4 * 4;  // float offset in lw
        async_ld128(&lw[0][0] + fo, wsrc + (size_t)j * DD + c4 * 4);
      }
    }
    wait_async0();
    __syncthreads();  // all waves' staged data visible

    v8f ir = {0.f, 0.f, 0.f, 0.f, 0.f, 0.f, 0.f, 0.f};
    v8f iz = ir, inn = ir, hr = ir, hz = ir, hn = ir;
#pragma unroll
    for (int kt = 0; kt < 16; ++kt) {
      const int k0 = kt * 4 + kb;
      v2f b;
#define LOADB(s, g)                                                     \
  {                                                                     \
    const float2 t =                                                    \
        *(const float2*)&lw[s][((g)*16 + n) * LROW + k0];               \
    b.x = t.x;                                                          \
    b.y = t.y;                                                          \
  }
      LOADB(0, 0); ir  = wmma4(aA[kt], b, ir);
      LOADB(0, 1); iz  = wmma4(aA[kt], b, iz);
      LOADB(0, 2); inn = wmma4(aA[kt], b, inn);
      LOADB(1, 0); hr  = wmma4(aH[kt], b, hr);
      LOADB(1, 1); hz  = wmma4(aH[kt], b, hz);
      LOADB(1, 2); hn  = wmma4(aH[kt], b, hn);
#undef LOADB
    }

    const float bir = b_ih[dt * 16 + n];
    const float biz = b_ih[DD + dt * 16 + n];
    const float bin = b_ih[2 * DD + dt * 16 + n];
    const float bhr = b_hh[dt * 16 + n];
    const float bhz = b_hh[DD + dt * 16 + n];
    const float bhn = b_hh[2 * DD + dt * 16 + n];

    if (active) {  // wave-uniform: tail duplicates must not race the owner
#pragma unroll
      for (int i = 0; i < 8; ++i) {
        const int    row  = r0 + i + half * 8;
        const size_t idx  = (size_t)row * DD + dt * 16 + n;
        const float  hval = h[idx];
        const float  r  = 1.f / (1.f + __expf(-(ir[i] + bir + hr[i] + bhr)));
        const float  z  = 1.f / (1.f + __expf(-(iz[i] + biz + hz[i] + bhz)));
        const float  nn = tanhf(inn[i] + bin + r * (hn[i] + bhn));
        hout[idx] = (1.f - z) * nn + z * hval;
      }
    }
  }
}

// ---------------------------------------------------------------------------
extern "C" void kernel_launch(void* const* d_in, const int* in_sizes, int n_in,
                              void* d_out, int out_size, void* d_ws,
                              size_t ws_size, hipStream_t stream) {
  const float* x      = (const float*)d_in[0];
  const int*   ei     = (const int*)d_in[1];
  const float* weight = (const float*)d_in[2];  // [L][64][64]
  const float* w_ih   = (const float*)d_in[3];  // [192][64]
  const float* w_hh   = (const float*)d_in[4];  // [192][64]
  const float* b_ih   = (const float*)d_in[5];  // [192]
  const float* b_hh   = (const float*)d_in[6];  // [192]

  float* m_ws   = (float*)d_ws;                        // N*D floats
  float* agg_ws = m_ws + (size_t)NN * DD;              // N*D floats
  float* h_ws   = agg_ws + (size_t)NN * DD;            // N*D floats

  const int grid2 = (EE * 16) / 256;                   // 50000, exact

  for (int step = 0; step < LL; ++step) {
    const float* hin  = (step == 0) ? x : h_ws;
    float*       hout = (step == LL - 1) ? (float*)d_out : h_ws;
    const float* W    = weight + (size_t)step * DD * DD;

    ggnn_msg_gemm<<<GRID_T, 256, 0, stream>>>(hin, W, m_ws, agg_ws);
    ggnn_scatter<<<grid2, 256, 0, stream>>>(ei, m_ws, agg_ws);
    ggnn_gru<<<GRID_T, 256, 0, stream>>>(agg_ws, hin, w_ih, w_hh, b_ih, b_hh,
                                         hout);
  }
}